// SGCEvaluator_36438502539416
// MI455X (gfx1250) — compile-verified
//
#include <hip/hip_runtime.h>
#include <hip/hip_bf16.h>

#define D_FEAT 128
#define N_CLASS 40

typedef float v2f __attribute__((ext_vector_type(2)));
typedef float v8f __attribute__((ext_vector_type(8)));

// ---------------------------------------------------------------------------
// Kernel 1: weighted degree accumulation (segment sums via HW f32 atomics)
// ---------------------------------------------------------------------------
__global__ void sgc_deg_kernel(const int* __restrict__ src,
                               const int* __restrict__ dst,
                               const float* __restrict__ ew,
                               float* __restrict__ degO,
                               float* __restrict__ degI,
                               int nE) {
    int e = blockIdx.x * blockDim.x + threadIdx.x;
    if (e >= nE) return;
    float v = ew[e];
    unsafeAtomicAdd(&degO[src[e]], v);   // -> global_atomic_add_f32
    unsafeAtomicAdd(&degI[dst[e]], v);
}

// ---------------------------------------------------------------------------
// Kernel 2: EdgeWeightNorm('both'): w = ew * rsqrt(degO[src]) * rsqrt(degI[dst])
// ---------------------------------------------------------------------------
__global__ void sgc_norm_kernel(const int* __restrict__ src,
                                const int* __restrict__ dst,
                                const float* __restrict__ ew,
                                const float* __restrict__ degO,
                                const float* __restrict__ degI,
                                float* __restrict__ w,
                                int nE) {
    int e = blockIdx.x * blockDim.x + threadIdx.x;
    if (e >= nE) return;
    w[e] = ew[e] * __frsqrt_rn(degO[src[e]]) * __frsqrt_rn(degI[dst[e]]);
}

// ---------------------------------------------------------------------------
// Kernel 3: SpMM hop: hout[dst] += w_e * hin[src].  One wave per edge,
// each lane moves float4 (128 floats / 32 lanes), 4 HW f32 atomics per lane.
// h (51.2 MB) is L2-resident on MI455X (192 MB L2), so this is an L2
// gather/scatter kernel, not an HBM kernel.
// ---------------------------------------------------------------------------
__global__ void sgc_spmm_kernel(const float4* __restrict__ hin,
                                const int* __restrict__ src,
                                const int* __restrict__ dst,
                                const float* __restrict__ w,
                                float* __restrict__ hout,
                                int nE) {
    int e = blockIdx.x * (blockDim.x >> 5) + (threadIdx.x >> 5);
    if (e >= nE) return;                       // wave-uniform exit
    int lane = threadIdx.x & 31;
    int s = src[e];
    int d = dst[e];
    float we = w[e];
    float4 v = hin[(size_t)s * (D_FEAT / 4) + lane];
    float* o = hout + (size_t)d * D_FEAT + lane * 4;
    unsafeAtomicAdd(o + 0, v.x * we);
    unsafeAtomicAdd(o + 1, v.y * we);
    unsafeAtomicAdd(o + 2, v.z * we);
    unsafeAtomicAdd(o + 3, v.w * we);
}

// ---------------------------------------------------------------------------
// Kernel 4: fc projection out = h @ W + b using V_WMMA_F32_16X16X4_F32.
// One wave computes a 16x16 output tile; K=128 -> 32 wmma steps of K=4.
// f32 WMMA preserves the reference's f32 math exactly.
//
// A 16x4 f32 layout : lanes 0-15 hold K = k+0,k+1 (v0,v1);
//                     lanes 16-31 hold K = k+2,k+3.
// C/D 16x16 f32     : VGPR r -> M = r + 8*(lane>=16), N = lane&15.
// EXEC must be all-ones at the WMMA: no divergent branches in the k-loop
// (OOB columns handled with clamped addresses + multiplicative mask).
// ---------------------------------------------------------------------------
__global__ void sgc_fc_wmma_kernel(const float* __restrict__ h,
                                   const float* __restrict__ W,
                                   const float* __restrict__ bias,
                                   float* __restrict__ out,
                                   int n) {
    const int colTiles = (N_CLASS + 15) / 16;              // 3
    int wavesPerBlock = blockDim.x >> 5;
    int waveId = blockIdx.x * wavesPerBlock + (threadIdx.x >> 5);
    int nRowTiles = (n + 15) / 16;
    int rowTile = waveId / colTiles;
    int colTile = waveId % colTiles;
    if (rowTile >= nRowTiles) return;                      // wave-uniform exit

    int lane = threadIdx.x & 31;
    int m = lane & 15;
    int kHalf = (lane >> 4) << 1;                          // 0 or 2
    int row0 = rowTile * 16;
    int col0 = colTile * 16;
    int col = col0 + m;

    // Clamp (no divergence) instead of branching: keeps EXEC all-ones.
    int arow = row0 + m;
    if (arow > n - 1) arow = n - 1;
    int colC = (col < N_CLASS) ? col : (N_CLASS - 1);
    float colMask = (col < N_CLASS) ? 1.0f : 0.0f;

    const float* aPtr = h + (size_t)arow * D_FEAT + kHalf;
    const float* bPtr = W + (size_t)kHalf * N_CLASS + colC;

    v8f acc = {};
    #pragma unroll
    for (int k = 0; k < D_FEAT; k += 4) {
        v2f a, b;
        a.x = aPtr[0];
        a.y = aPtr[1];
        b.x = bPtr[0] * colMask;
        b.y = bPtr[N_CLASS] * colMask;
        acc = __builtin_amdgcn_wmma_f32_16x16x4_f32(
            /*neg_a=*/false, a, /*neg_b=*/false, b,
            /*c_mod=*/(short)0, acc, /*reuse_a=*/false, /*reuse_b=*/false);
        aPtr += 4;
        bPtr += 4 * N_CLASS;
    }

    float bv = bias[colC];
    int mBase = row0 + ((lane >> 4) << 3);                 // +8 for upper half
    #pragma unroll
    for (int r = 0; r < 8; ++r) {
        int rr = mBase + r;
        if (col < N_CLASS && rr < n)
            out[(size_t)rr * N_CLASS + col] = acc[r] + bv;
    }
}

// ---------------------------------------------------------------------------
// Launch
// ---------------------------------------------------------------------------
extern "C" void kernel_launch(void* const* d_in, const int* in_sizes, int n_in,
                              void* d_out, int out_size, void* d_ws, size_t ws_size,
                              hipStream_t stream) {
    const float* features = (const float*)d_in[0];   // [N, 128]
    const int*   src      = (const int*)d_in[1];     // [E]
    const int*   dst      = (const int*)d_in[2];     // [E]
    const float* ew       = (const float*)d_in[3];   // [E]
    const float* W        = (const float*)d_in[4];   // [128, 40]
    const float* bias     = (const float*)d_in[5];   // [40]

    int N = in_sizes[0] / D_FEAT;
    int E = in_sizes[1];

    // Workspace layout (floats): degO[N] | degI[N] | w[E] | h1[N*D] | h2[N*D]
    float* degO = (float*)d_ws;
    float* degI = degO + N;
    float* wn   = degI + N;
    float* h1   = wn + E;
    float* h2   = h1 + (size_t)N * D_FEAT;

    // Zero accumulators (memset nodes are graph-capturable).
    hipMemsetAsync(degO, 0, (size_t)2 * N * sizeof(float), stream);
    hipMemsetAsync(h1, 0, (size_t)N * D_FEAT * sizeof(float), stream);
    hipMemsetAsync(h2, 0, (size_t)N * D_FEAT * sizeof(float), stream);

    // 1) weighted degrees
    {
        int threads = 256;
        int blocks = (E + threads - 1) / threads;
        sgc_deg_kernel<<<blocks, threads, 0, stream>>>(src, dst, ew, degO, degI, E);
    }
    // 2) edge weight normalization
    {
        int threads = 256;
        int blocks = (E + threads - 1) / threads;
        sgc_norm_kernel<<<blocks, threads, 0, stream>>>(src, dst, ew, degO, degI, wn, E);
    }
    // 3) two SpMM hops (wave per edge)
    {
        int threads = 256;                       // 8 waves/block
        int blocks = (E + 7) / 8;
        sgc_spmm_kernel<<<blocks, threads, 0, stream>>>(
            (const float4*)features, src, dst, wn, h1, E);
        sgc_spmm_kernel<<<blocks, threads, 0, stream>>>(
            (const float4*)h1, src, dst, wn, h2, E);
    }
    // 4) fc projection via f32 WMMA
    {
        int colTiles = (N_CLASS + 15) / 16;      // 3
        int nRowTiles = (N + 15) / 16;
        long totalWaves = (long)nRowTiles * colTiles;
        int threads = 256;                       // 8 waves/block
        int blocks = (int)((totalWaves + 7) / 8);
        sgc_fc_wmma_kernel<<<blocks, threads, 0, stream>>>(h2, W, bias, (float*)d_out, N);
    }
}